// YOLOLayer_19816979104601
// MI455X (gfx1250) — compile-verified
//
#include <hip/hip_runtime.h>
#include <cstdint>

// ---------------------------------------------------------------------------
// YOLO head decode for MI455X (gfx1250).
// Memory-bound transpose+elementwise: 377 MB traffic -> ~16us floor @23.3TB/s.
// Strategy: TDM (tensor_load_to_lds) stages [85 x 127] channel-major tiles
// into LDS (stride 127 dwords = odd -> conflict-free transposed reads on the
// 64-bank LDS), then coalesced [127 x 85] stores. One v_exp_f32 + one
// v_rcp_f32 per element (shared exp2 for sigmoid/exp lanes via sign select).
// ---------------------------------------------------------------------------

#define GRID_G   76
#define GG       (GRID_G * GRID_G)   // 5776
#define DD       85                  // 5 + 80 classes
#define AA       3
#define BATCH    32
#define PT       127                 // pixels per full tile (odd -> no LDS bank conflicts)
#define NTILES   46                  // 45 full tiles + tail
#define TAILP    61                  // 5776 - 45*127
#define LOG2E    1.4426950408889634f
#define STRIDEF  8.0f

typedef unsigned int u32;
typedef __attribute__((ext_vector_type(4))) u32 u32x4;
typedef __attribute__((ext_vector_type(8))) int i32x8;
typedef __attribute__((ext_vector_type(4))) int i32x4;

#if defined(__HIP_DEVICE_COMPILE__) && __has_builtin(__builtin_amdgcn_tensor_load_to_lds)
#define USE_TDM 1
#else
#define USE_TDM 0
#endif

__device__ __constant__ float kAnchors[AA][2] = {
    {10.0f, 13.0f}, {16.0f, 30.0f}, {33.0f, 23.0f}};

__device__ inline float fast_exp2(float x) {
#if defined(__HIP_DEVICE_COMPILE__) && __has_builtin(__builtin_amdgcn_exp2f)
  return __builtin_amdgcn_exp2f(x);     // v_exp_f32
#else
  return exp2f(x);
#endif
}

__device__ inline float fast_rcp(float x) {
#if defined(__HIP_DEVICE_COMPILE__) && __has_builtin(__builtin_amdgcn_rcpf)
  return __builtin_amdgcn_rcpf(x);      // v_rcp_f32
#else
  return 1.0f / x;
#endif
}

__global__ __launch_bounds__(256) void yolo_decode_kernel(
    const float* __restrict__ x, float* __restrict__ out) {
  __shared__ float tile[DD * PT];       // 43,180 B of the 320 KB WGP LDS

  const int tid = threadIdx.x;
  const int t   = blockIdx.x;           // tile index within one (b,a) image
  const int ba  = blockIdx.y;           // b*3 + a
  const int b   = ba / AA;
  const int a   = ba - AA * b;
  const int c0  = t * PT;                          // first pixel of tile
  const int Pt  = (t == NTILES - 1) ? TAILP : PT;  // pixels in this tile
  const int chanBase = b * (AA * DD) + a * DD;     // input channel of d==0

#if USE_TDM
  {
    // Each of the 8 waves DMAs its own slice of rows (TDM ignores EXEC, so
    // per-wave issue avoids any predication hazard and parallelizes the DMA).
    const int w    = tid >> 5;                       // wave id 0..7
    const int r0   = (w < 5) ? w * 11 : 55 + (w - 5) * 10;
    const int rows = (w < 5) ? 11 : 10;              // 5*11 + 3*10 = 85

    const uint64_t gaddr = (uint64_t)(uintptr_t)x +
        4ull * ((uint64_t)(chanBase + r0) * GG + (uint64_t)c0);
    const u32 lds_off =
        (u32)(uintptr_t)(&tile[0]) + (u32)(r0 * Pt * 4);

    // ---- D# group 0 (128b): count | lds_addr | global_addr | type=2 ----
    u32x4 g0;
    g0.x = 1u;                                        // count=1 (valid user D#)
    g0.y = lds_off;                                   // lds_addr (bytes)
    g0.z = (u32)(gaddr & 0xFFFFFFFFull);              // global_addr[31:0]
    g0.w = (u32)((gaddr >> 32) & 0x01FFFFFFu)         // global_addr[56:32]
           | 0x80000000u;                             // type = 2 ("image")

    // ---- D# group 1 (256b): 2-D tile, data_size=4B, no pad/iterate ----
    i32x8 g1;
    g1[0] = (int)(2u << 16);                          // data_size = 4 bytes
    g1[1] = (int)(((u32)Pt & 0xFFFFu) << 16);         // tensor_dim0 = Pt (tile==tensor, no OOB)
    g1[2] = (int)(((u32)rows & 0xFFFFu) << 16);       // tensor_dim1 = rows
    g1[3] = (int)(((u32)Pt & 0xFFFFu) << 16);         // tile_dim0 = Pt
    g1[4] = rows;                                     // tile_dim1 = rows, tile_dim2 = 0
    g1[5] = GG;                                       // tensor_dim0_stride = 5776
    g1[6] = 0;
    g1[7] = 0;

    i32x4 gz4 = {0, 0, 0, 0};                         // groups 2/3 unused (2-D)
    i32x8 gz8 = {0, 0, 0, 0, 0, 0, 0, 0};             // trailing group (unused)
    __builtin_amdgcn_tensor_load_to_lds(g0, g1, gz4, gz4, gz8, 0);
    __builtin_amdgcn_s_wait_tensorcnt(0);             // wave's DMA complete
  }
#else
  // Fallback: cooperative coalesced loads -> LDS (same layout).
  for (int idx = tid; idx < DD * Pt; idx += 256) {
    int d = idx / Pt;
    int p = idx - d * Pt;
    tile[d * Pt + p] =
        x[(size_t)(chanBase + d) * GG + (size_t)(c0 + p)];
  }
#endif
  __syncthreads();   // all waves' slices visible before transposed reads

  const float aw = kAnchors[a][0];
  const float ah = kAnchors[a][1];
  const int nElem = DD * Pt;
  float* outBase = out + (size_t)((size_t)ba * GG + (size_t)c0) * DD;

  for (int e = tid; e < nElem; e += 256) {
    const int p = e / DD;                // pixel within tile
    const int d = e - DD * p;            // output feature index
    const float v = tile[d * Pt + p];    // stride-127 read: conflict-free

    const int pix = c0 + p;
    const int gi  = pix / GRID_G;        // 'ij' meshgrid: d==1 -> +i
    const int gj  = pix - GRID_G * gi;   //               d==2 -> +j

    const bool isexp = (d == 3) || (d == 4);
    // One shared v_exp_f32: exp lanes use +log2e*v, sigmoid lanes -log2e*v.
    const float t2 = fast_exp2(v * (isexp ? LOG2E : -LOG2E));
    const float u  = isexp ? t2 : fast_rcp(1.0f + t2);   // exp(v) or sigmoid(v)

    const float bias  = (d == 1) ? (float)gi : (d == 2) ? (float)gj : 0.0f;
    const float scale = (d == 1 || d == 2) ? STRIDEF
                      : (d == 3)           ? aw
                      : (d == 4)           ? ah
                                           : 1.0f;
    outBase[e] = (u + bias) * scale;     // fully coalesced b32 stores
  }
}

extern "C" void kernel_launch(void* const* d_in, const int* in_sizes, int n_in,
                              void* d_out, int out_size, void* d_ws, size_t ws_size,
                              hipStream_t stream) {
  const float* x = (const float*)d_in[0];
  float* out = (float*)d_out;
  dim3 grid(NTILES, BATCH * AA);   // 46 x 96 = 4416 workgroups
  yolo_decode_kernel<<<grid, 256, 0, stream>>>(x, out);
}